// CDLoss_31980326486602
// MI455X (gfx1250) — compile-verified
//
#include <hip/hip_runtime.h>
#include <hip/hip_bf16.h>

typedef __attribute__((ext_vector_type(2))) float v2f;
typedef __attribute__((ext_vector_type(8))) float v8f;

#define TPB 256
#define WAVES 8
#define ROWS_PER_WAVE 16
#define ROWS_PER_WG (WAVES * ROWS_PER_WAVE)   // 128 rows of P per workgroup
#define QCHUNK 512                            // Q points staged in LDS per outer step

// For each point in P, find min squared distance to any point in Q (row mins),
// sum them per workgroup -> partials[blockIdx.x].
// Two "sides" packed into one grid: blocks [0, nbSide) do (pcs1 -> pcs2),
// blocks [nbSide, 2*nbSide) do (pcs2 -> pcs1).
//
// WMMA mapping per 16x16 tile (V_WMMA_F32_16X16X4_F32, K padded 3->4):
//   A(16x4)  row i : (px, py, pz, 1)          -- loop invariant, in registers
//   B(4x16)  col j : (-2qx, -2qy, -2qz, ||q||^2)
//   C = 0       =>  D_ij = ||q_j||^2 - 2 p_i . q_j
// ||p_i||^2 added (and clamp to 0 applied) after the min reduction since both
// commute with min.
//
// The B operand for each Q point is PRE-PACKED into LDS as an aligned float4
// (-2qx, -2qy, -2qz, sq2) during the staging phase, so the hot loop's operand
// build is a single ds_load_b64: lanes 0-15 read bytes [0,8) (K0,K1), lanes
// 16-31 read bytes [8,16) (K2,K3) -- exactly the mirrored WMMA B layout
// (VGPR0 = K0|K2, VGPR1 = K1|K3 across lane halves).
__global__ __launch_bounds__(TPB) void chamfer_rowmin_kernel(
    const float* __restrict__ pcs1, const float* __restrict__ pcs2,
    float* __restrict__ partials, int Bn, int Nn, int Mn, int nbSide)
{
    __shared__ float4 lds_b[QCHUNK];        // 8 KB of the 320 KB/WGP LDS
    __shared__ float lds_part[WAVES];

    const int tid  = threadIdx.x;
    const int wave = tid >> 5;
    const int lane = tid & 31;
    const int col  = lane & 15;             // tile column
    const int hi   = lane >> 4;             // 0: K0/K1 half, 1: K2/K3 half

    const int gid  = blockIdx.x;
    const int side = (gid >= nbSide) ? 1 : 0;
    const int lb   = gid - side * nbSide;

    const float* __restrict__ P = side ? pcs2 : pcs1;
    const float* __restrict__ Q = side ? pcs1 : pcs2;

    const int rowBlocks = Nn / ROWS_PER_WG;
    const int batch = lb / rowBlocks;
    const int rb    = lb % rowBlocks;

    const float* Pb = P + (size_t)batch * Nn * 3;
    const float* Qb = Q + (size_t)batch * Mn * 3;

    // ---- Build loop-invariant A operand (16x4 f32 -> 2 VGPRs) ----
    // ISA layout: VGPR0 = K0 (lanes 0-15) | K2 (lanes 16-31)
    //             VGPR1 = K1 (lanes 0-15) | K3 (lanes 16-31)
    const int row = rb * ROWS_PER_WG + wave * ROWS_PER_WAVE + col;
    const float px = Pb[row * 3 + 0];
    const float py = Pb[row * 3 + 1];
    const float pz = Pb[row * 3 + 2];
    const float sq1 = px * px + py * py + pz * pz;

    v2f amat;
    amat.x = hi ? pz   : px;   // K2 : K0
    amat.y = hi ? 1.0f : py;   // K3 : K1

    v8f czero = {};            // C = 0

    float acc[8];
#pragma unroll
    for (int r = 0; r < 8; ++r) acc[r] = 3.4e38f;

    // Per-lane base into the packed B table: point stride = 16B, half = 8B.
    const float2* __restrict__ bbase =
        reinterpret_cast<const float2*>(lds_b) + (col * 2 + hi);

    for (int m0 = 0; m0 < Mn; m0 += QCHUNK) {
        __syncthreads();
        // Stage + pre-pack QCHUNK points: each thread packs QCHUNK/TPB points.
#pragma unroll
        for (int j = tid; j < QCHUNK; j += TPB) {
            const float qx = Qb[(size_t)(m0 + j) * 3 + 0];
            const float qy = Qb[(size_t)(m0 + j) * 3 + 1];
            const float qz = Qb[(size_t)(m0 + j) * 3 + 2];
            const float sq2 = qx * qx + qy * qy + qz * qz;
            lds_b[j] = make_float4(-2.0f * qx, -2.0f * qy, -2.0f * qz, sq2);
        }
        __syncthreads();

        // Prefetch next chunk into cache while we compute this one.
        if (m0 + QCHUNK < Mn)
            __builtin_prefetch(Qb + (size_t)(m0 + QCHUNK) * 3 + tid * 6, 0, 1);

#pragma unroll 8
        for (int s = 0; s < QCHUNK; s += 16) {
            // Single b64 LDS read = this lane's half of the B operand.
            const float2 bv = bbase[s * 2];
            v2f bmat;
            bmat.x = bv.x;
            bmat.y = bv.y;

            v8f d = __builtin_amdgcn_wmma_f32_16x16x4_f32(
                false, amat, false, bmat, (short)0, czero, false, false);

            // Running per-row minimum. D layout: VGPR r = row r (lanes 0-15)
            // and row r+8 (lanes 16-31); lane%16 = column.
#pragma unroll
            for (int r = 0; r < 8; ++r)
                acc[r] = fminf(acc[r], d[r]);
        }
    }

    // ---- Min-reduce across the 16 lanes of each half (columns) ----
#pragma unroll
    for (int off = 1; off < 16; off <<= 1) {
#pragma unroll
        for (int r = 0; r < 8; ++r)
            acc[r] = fminf(acc[r], __shfl_xor(acc[r], off, 32));
    }

    // acc[r] now = min over all columns for tile-row (hi? r+8 : r).
    // Fetch that row's ||p||^2 (it lives in lane (hi? r+8 : r) of this wave),
    // add it, clamp, and sum the 8 rows held by this half.
    float s8 = 0.0f;
#pragma unroll
    for (int r = 0; r < 8; ++r) {
        const float sq1_r = __shfl(sq1, hi ? (r + 8) : r, 32);
        s8 += fmaxf(acc[r] + sq1_r, 0.0f);
    }
    // Combine the two halves (rows 0-7 + rows 8-15).
    s8 += __shfl_xor(s8, 16, 32);

    if (lane == 0) lds_part[wave] = s8;
    __syncthreads();
    if (tid == 0) {
        float t = 0.0f;
#pragma unroll
        for (int w = 0; w < WAVES; ++w) t += lds_part[w];
        partials[gid] = t;
    }
}

// Deterministic fixed-order reduction of the per-workgroup partials.
__global__ __launch_bounds__(TPB) void chamfer_finalize_kernel(
    const float* __restrict__ partials, int n1, int n2,
    float inv1, float inv2, float* __restrict__ out)
{
    __shared__ float s1[TPB];
    __shared__ float s2[TPB];
    const int t = threadIdx.x;
    float a = 0.0f, b = 0.0f;
    for (int i = t; i < n1; i += TPB) a += partials[i];
    for (int i = t; i < n2; i += TPB) b += partials[n1 + i];
    s1[t] = a; s2[t] = b;
    __syncthreads();
    for (int off = TPB / 2; off > 0; off >>= 1) {
        if (t < off) { s1[t] += s1[t + off]; s2[t] += s2[t + off]; }
        __syncthreads();
    }
    if (t == 0) out[0] = s1[0] * inv1 + s2[0] * inv2;
}

extern "C" void kernel_launch(void* const* d_in, const int* in_sizes, int n_in,
                              void* d_out, int out_size, void* d_ws, size_t ws_size,
                              hipStream_t stream) {
    const float* pcs1 = (const float*)d_in[0];   // [8, 8192, 3] f32
    const float* pcs2 = (const float*)d_in[1];   // [8, 8192, 3] f32
    float* out = (float*)d_out;                  // scalar f32
    float* partials = (float*)d_ws;              // 2 * 512 floats

    constexpr int B = 8, N = 8192, M = 8192;
    constexpr int nbSide = B * (N / ROWS_PER_WG);   // 512 workgroups per side

    chamfer_rowmin_kernel<<<2 * nbSide, TPB, 0, stream>>>(
        pcs1, pcs2, partials, B, N, M, nbSide);
    chamfer_finalize_kernel<<<1, TPB, 0, stream>>>(
        partials, nbSide, nbSide, 1.0f / ((float)B * N), 1.0f / ((float)B * M), out);
}